// BatchedVariableGRN_54176717471838
// MI455X (gfx1250) — compile-verified
//
#include <hip/hip_runtime.h>

typedef float v2f __attribute__((ext_vector_type(2)));
typedef float v8f __attribute__((ext_vector_type(8)));

#define N_  2048
#define V_  204
#define H_  160
#define G_  320     // 2*H
#define MT  64      // rows per workgroup
#define NWAVES 16
#define NTHR 512
#define PH  162     // LDS pitch for H1/H2/Y tiles (bank-conflict padding)
#define PB  321     // LDS pitch for transposed glu_w block
#define KB  16      // k-rows of glu_w staged per block
#define LN_EPS 1e-5f

__global__ __launch_bounds__(NTHR)
void grn_fused_kernel(const float* __restrict__ x,
                      const float* __restrict__ fc1_w, const float* __restrict__ fc1_b,
                      const float* __restrict__ fc2_w, const float* __restrict__ fc2_b,
                      const float* __restrict__ glu_w, const float* __restrict__ glu_b,
                      const float* __restrict__ skip_w, const float* __restrict__ skip_b,
                      const float* __restrict__ ln_g, const float* __restrict__ ln_b,
                      float* __restrict__ out)
{
    __shared__ float sH1[MT * PH];   // elu(fc1) tile; later reused as Y tile
    __shared__ float sH2[MT * PH];   // fc2 output tile
    __shared__ float sBT[KB * PB];   // transposed glu_w k-block: sBT[k][g]
    __shared__ float sXS[MT];        // per-row scalar x

    const int v    = blockIdx.y;
    const int n0   = blockIdx.x * MT;
    const int tid  = threadIdx.x;
    const int wave = tid >> 5;
    const int lane = tid & 31;
    const int lhalf = lane >> 4;     // 0/1 : which half of the wave
    const int l16   = lane & 15;

    // ---------------- Stage A: h1 = elu(xs * fc1_w + fc1_b), stash xs -------
    if (tid < MT) sXS[tid] = x[(size_t)(n0 + tid) * V_ + v];
    for (int idx = tid; idx < MT * H_; idx += NTHR) {
        int m = idx / H_, h = idx - m * H_;
        float xs = x[(size_t)(n0 + m) * V_ + v];
        float z  = xs * fc1_w[v * H_ + h] + fc1_b[v * H_ + h];
        sH1[m * PH + h] = (z > 0.f) ? z : (__expf(z) - 1.f);
    }
    __syncthreads();

    // ---------------- Stage B: H2 = H1 @ fc2_w[v] + fc2_b[v] ----------------
    // 4 m-tiles x 10 col-tiles = 40 WMMA output tiles over 16 waves
    for (int t = wave; t < (MT / 16) * (H_ / 16); t += NWAVES) {
        int mt = t / (H_ / 16), ct = t - mt * (H_ / 16);
        float bias = fc2_b[v * H_ + ct * 16 + l16];
        v8f acc;
        #pragma unroll
        for (int r = 0; r < 8; ++r) acc[r] = bias;

        const float* Bb = fc2_w + (size_t)v * H_ * H_ + ct * 16 + l16;
        const float* Ab = sH1 + (mt * 16 + l16) * PH;
        for (int k0 = 0; k0 < H_; k0 += 4) {
            int kk = k0 + lhalf * 2;
            v2f a; a.x = Ab[kk];               a.y = Ab[kk + 1];
            v2f b; b.x = Bb[(size_t)kk * H_];  b.y = Bb[(size_t)(kk + 1) * H_];
            acc = __builtin_amdgcn_wmma_f32_16x16x4_f32(false, a, false, b,
                                                        (short)0, acc, false, false);
        }
        float* dst = sH2 + ct * 16 + l16;
        #pragma unroll
        for (int r = 0; r < 8; ++r) dst[(mt * 16 + lhalf * 8 + r) * PH] = acc[r];
    }
    __syncthreads();

    // ---------------- Stage C: GLU = H2 @ glu_w[v]^T + glu_b ----------------
    // Each wave owns (fc, gate) tile PAIRS: pair p -> g-tiles (ht, ht+10).
    // pairs p = wave, wave+16, wave+32(if wave<8)  (40 pairs total)
    const int havep2 = (wave < 8);
    int mts[3], hts[3];
    #pragma unroll
    for (int q = 0; q < 3; ++q) {
        int p = wave + 16 * q;
        mts[q] = p / 10;
        hts[q] = p - (p / 10) * 10;
    }
    v8f accF[3], accG[3];
    #pragma unroll
    for (int q = 0; q < 3; ++q) {
        if (q == 2 && !havep2) continue;
        float bF = glu_b[v * G_ + hts[q] * 16 + l16];
        float bG = glu_b[v * G_ + (hts[q] + 10) * 16 + l16];
        #pragma unroll
        for (int r = 0; r < 8; ++r) { accF[q][r] = bF; accG[q][r] = bG; }
    }

    for (int kb = 0; kb < H_; kb += KB) {
        __syncthreads();               // previous sBT block fully consumed
        // stage transposed glu_w block: sBT[k][g] = glu_w[v, g, kb+k]
        for (int idx = tid; idx < KB * G_; idx += NTHR) {
            int g = idx >> 4, kk = idx & 15;
            sBT[kk * PB + g] = glu_w[((size_t)(v * G_ + g)) * H_ + kb + kk];
        }
        __syncthreads();

        #pragma unroll
        for (int q = 0; q < 3; ++q) {
            if (q == 2 && !havep2) continue;
            const float* Ab = sH2 + (mts[q] * 16 + l16) * PH + kb;
            const int gF = hts[q] * 16 + l16;
            const int gG = (hts[q] + 10) * 16 + l16;
            #pragma unroll
            for (int k0 = 0; k0 < KB; k0 += 4) {
                int kk = k0 + lhalf * 2;
                v2f a;  a.x  = Ab[kk];                 a.y  = Ab[kk + 1];
                v2f bf; bf.x = sBT[kk * PB + gF];      bf.y = sBT[(kk + 1) * PB + gF];
                v2f bg; bg.x = sBT[kk * PB + gG];      bg.y = sBT[(kk + 1) * PB + gG];
                accF[q] = __builtin_amdgcn_wmma_f32_16x16x4_f32(false, a, false, bf,
                                                                (short)0, accF[q], false, false);
                accG[q] = __builtin_amdgcn_wmma_f32_16x16x4_f32(false, a, false, bg,
                                                                (short)0, accG[q], false, false);
            }
        }
    }

    // ---------------- Stage D: y = skip + fc*sigmoid(gate) -> LDS -----------
    float* sY = sH1;                   // sH1 is dead; reuse as Y tile
    #pragma unroll
    for (int q = 0; q < 3; ++q) {
        if (q == 2 && !havep2) continue;
        int h = hts[q] * 16 + l16;
        float sw = skip_w[v * H_ + h];
        float sb = skip_b[v * H_ + h];
        #pragma unroll
        for (int r = 0; r < 8; ++r) {
            int m = mts[q] * 16 + lhalf * 8 + r;
            float sig = 1.f / (1.f + __expf(-accG[q][r]));
            float yv  = sXS[m] * sw + sb + accF[q][r] * sig;
            sY[m * PH + h] = yv;
        }
    }
    __syncthreads();

    // ---------------- Stage E: LayerNorm over H, write output ---------------
    // wave w handles rows 4w .. 4w+3; lanes cover 160 cols (5 each)
    for (int rr = 0; rr < 4; ++rr) {
        int m = wave * 4 + rr;
        float vals[5];
        float s = 0.f, s2 = 0.f;
        #pragma unroll
        for (int j = 0; j < 5; ++j) {
            float y = sY[m * PH + lane + 32 * j];
            vals[j] = y; s += y; s2 += y * y;
        }
        #pragma unroll
        for (int off = 16; off > 0; off >>= 1) {
            s  += __shfl_xor(s,  off, 32);
            s2 += __shfl_xor(s2, off, 32);
        }
        float mu  = s * (1.f / H_);
        float var = s2 * (1.f / H_) - mu * mu;
        float rs  = rsqrtf(var + LN_EPS);
        size_t ob = ((size_t)(n0 + m) * V_ + v) * H_;
        #pragma unroll
        for (int j = 0; j < 5; ++j) {
            int h = lane + 32 * j;
            out[ob + h] = (vals[j] - mu) * rs * ln_g[h] + ln_b[h];
        }
    }
}

extern "C" void kernel_launch(void* const* d_in, const int* in_sizes, int n_in,
                              void* d_out, int out_size, void* d_ws, size_t ws_size,
                              hipStream_t stream) {
    const float* x      = (const float*)d_in[0];
    const float* fc1_w  = (const float*)d_in[1];
    const float* fc1_b  = (const float*)d_in[2];
    const float* fc2_w  = (const float*)d_in[3];
    const float* fc2_b  = (const float*)d_in[4];
    const float* glu_w  = (const float*)d_in[5];
    const float* glu_b  = (const float*)d_in[6];
    const float* skip_w = (const float*)d_in[7];
    const float* skip_b = (const float*)d_in[8];
    const float* ln_g   = (const float*)d_in[9];
    const float* ln_b   = (const float*)d_in[10];

    dim3 grid(N_ / MT, V_);
    grn_fused_kernel<<<grid, NTHR, 0, stream>>>(x, fc1_w, fc1_b, fc2_w, fc2_b,
                                                glu_w, glu_b, skip_w, skip_b,
                                                ln_g, ln_b, (float*)d_out);
}